// SimpleRNN_24343874633926
// MI455X (gfx1250) — compile-verified
//
#include <hip/hip_runtime.h>
#include <math.h>

// ---------------------------------------------------------------------------
// LSTM (B=64, S=2048, WIN=50, E=128, H=256) for MI455X / gfx1250.
//  - fp32 WMMA (v_wmma_f32_16x16x4_f32) for all GEMMs (fp32-faithful).
//  - Recurrent scan: 16-WG workgroup cluster, W_hh weight-stationary in LDS
//    (64KB/WG slice, staged via TDM tensor_load_to_lds), per-step h exchange
//    through L2 ping-pong buffers filled with GLOBAL_LOAD_ASYNC_TO_LDS_B128,
//    hardware cluster barrier (s_barrier_signal/-wait -3) per step.
// ---------------------------------------------------------------------------

typedef float v2f __attribute__((ext_vector_type(2)));
typedef float v8f __attribute__((ext_vector_type(8)));
typedef unsigned int u32x4 __attribute__((ext_vector_type(4)));
typedef int i32x4 __attribute__((ext_vector_type(4)));
typedef int i32x8 __attribute__((ext_vector_type(8)));

// 8-arg form: (neg_a, A, neg_b, B, c_mod, C, reuse_a, reuse_b)
#define WMMA_F32(a, b, c) \
  __builtin_amdgcn_wmma_f32_16x16x4_f32(false, (a), false, (b), (short)0, (c), false, false)

#if defined(__has_attribute)
#if __has_attribute(amdgpu_cluster_dims)
#define CLUSTER_ATTR __attribute__((amdgpu_cluster_dims(16, 1, 1)))
#endif
#endif
#ifndef CLUSTER_ATTR
#define CLUSTER_ATTR
#endif

#if defined(__has_builtin)
#if __has_builtin(__builtin_amdgcn_tensor_load_to_lds) && \
    __has_builtin(__builtin_amdgcn_s_wait_tensorcnt)
#define HAS_TDM 1
#endif
#endif
#ifndef HAS_TDM
#define HAS_TDM 0
#endif

constexpr int B = 64, S = 2048, WIN = 50, E = 128, H = 256, G = 4 * H; // G=1024

__device__ __forceinline__ float sigmoidf_(float x) {
  return 1.0f / (1.0f + __expf(-x));
}

// ---------------------------------------------------------------------------
// Phase A: x_gates[s][b][g] = W_ih @ relu(W_emb @ pw[b,s,:] + b_emb) + b_ih+b_hh
// One workgroup (512 thr = 16 waves) per sequence position s.
// ---------------------------------------------------------------------------
__global__ __launch_bounds__(512) void xgates_kernel(
    const float* __restrict__ pw,   const float* __restrict__ Wemb,
    const float* __restrict__ bemb, const float* __restrict__ Wih,
    const float* __restrict__ bih,  const float* __restrict__ bhh,
    float* __restrict__ xg) {
  __shared__ float sPW[64 * 52];    // window tile, K padded 50->52 with zeros
  __shared__ float sWE[128 * 52];   // W_emb as [N=E][K], zero padded
  __shared__ float sEMB[64 * 132];  // emb tile [b][e], stride 132

  const int s = blockIdx.x;
  const int tid = threadIdx.x;
  const int lane = tid & 31, half = lane >> 4, l15 = lane & 15;  // wave32
  const int t = tid >> 5;                                        // wave id 0..15

  for (int i = tid; i < 64 * 52; i += 512) {
    int b = i / 52, k = i - b * 52;
    sPW[i] = (k < WIN) ? pw[((size_t)b * S + s) * WIN + k] : 0.0f;
  }
  for (int i = tid; i < 128 * 52; i += 512) {
    int n = i / 52, k = i - n * 52;
    sWE[i] = (k < WIN) ? Wemb[n * WIN + k] : 0.0f;
  }
  __syncthreads();

  // GEMM1: emb[64x128] = relu(PW[64x52] @ WembT[52x128] + b_emb)
#pragma unroll
  for (int it = 0; it < 2; ++it) {
    int tl = t + it * 16;
    int mb = tl & 3, nb = tl >> 2;
    v8f acc = {};
#pragma unroll
    for (int kc = 0; kc < 13; ++kc) {
      int k0 = kc * 4 + 2 * half;
      v2f af = *(const v2f*)&sPW[(mb * 16 + l15) * 52 + k0];
      v2f bf = *(const v2f*)&sWE[(nb * 16 + l15) * 52 + k0];
      acc = WMMA_F32(af, bf, acc);
    }
    int n = nb * 16 + l15;
    float be = bemb[n];
#pragma unroll
    for (int r = 0; r < 8; ++r) {
      int m = mb * 16 + r + 8 * half;
      sEMB[m * 132 + n] = fmaxf(acc[r] + be, 0.0f);
    }
  }
  __syncthreads();

  // GEMM2: xg[s][64x1024] = emb[64x128] @ WihT[128x1024] + (b_ih + b_hh)
#pragma unroll
  for (int j = 0; j < 4; ++j) {
    int gl = (t * 4 + j) * 16 + l15;  // output column for this lane
    float bias = bih[gl] + bhh[gl];
    v8f acc[4] = {};
    for (int k0 = 0; k0 < E; k0 += 4) {
      // W_ih is [G][E] == [N][K] storage: one b64 per B-fragment
      v2f bf = *(const v2f*)(Wih + (size_t)gl * E + k0 + 2 * half);
#pragma unroll
      for (int mb = 0; mb < 4; ++mb) {
        v2f af = *(const v2f*)&sEMB[(mb * 16 + l15) * 132 + k0 + 2 * half];
        acc[mb] = WMMA_F32(af, bf, acc[mb]);
      }
    }
#pragma unroll
    for (int mb = 0; mb < 4; ++mb) {
#pragma unroll
      for (int r = 0; r < 8; ++r) {
        int b = mb * 16 + r + 8 * half;
        xg[((size_t)s * B + b) * G + gl] = acc[mb][r] + bias;
      }
    }
  }
}

// ---------------------------------------------------------------------------
// Phase B: sequential LSTM scan. 16-workgroup cluster; WG w owns h-columns
// [w*16, w*16+16). W_hh slice (64x256 fp32 = 64KB) stays in LDS for all 2048
// steps (staged with the Tensor Data Mover). Per step: async-load full h
// (L2 ping-pong) -> LDS, WMMA gate slice, pointwise cell, publish h slice,
// cluster barrier.
// ---------------------------------------------------------------------------
CLUSTER_ATTR
__global__ __launch_bounds__(512) void lstm_kernel(
    const float* __restrict__ xg,  const float* __restrict__ Whh,
    const float* __restrict__ Wout, const float* __restrict__ c0,
    float* __restrict__ hglob, float* __restrict__ ypart) {
  __shared__ float sW[64 * 256];   // W_hh slice, [n_local=64][K=256]  (64KB)
  __shared__ float sH[64 * 256];   // full h_{s-1}, [b][k]             (64KB)
  __shared__ float sGT[64 * 64];   // gate slice [b][n_local]          (16KB)
  __shared__ float sC[64 * 16];    // private c slice                  (4KB)
  __shared__ float sHS[64 * 16];   // h*Wout staging                   (4KB)
  __shared__ float sWO[16];

  const int w = blockIdx.x;  // 0..15
  const int tid = threadIdx.x;
  const int lane = tid & 31, half = lane >> 4, l15 = lane & 15;
  const int t = tid >> 5;
  const int mb = t & 3;   // b-block of this wave's tile
  const int q = t >> 2;   // gate type (0=i,1=f,2=g,3=o)

  // ---- Stage W_hh slice: n_local = qq*16 + j  ->  g = qq*256 + w*16 + j ----
#if HAS_TDM
  if (tid < 32) {  // one elected wave drives 4 TDM 2D-tile descriptors
#pragma unroll
    for (int qq = 0; qq < 4; ++qq) {
      unsigned long long ga =
          (unsigned long long)(size_t)(Whh + ((size_t)(qq * H + w * 16)) * H);
      unsigned int la = (unsigned int)(size_t)&sW[qq * 16 * H];
      // D# group0: count=1 | lds_addr | global_addr[56:0] | type=2
      u32x4 g0 = {1u, la, (unsigned int)ga,
                  (unsigned int)((ga >> 32) & 0x01FFFFFFu) | (0x2u << 30)};
      // D# group1: data_size=4B; tensor_dim0=256; tensor_dim1=16;
      //            tile_dim0=256; tile_dim1=16; tensor_dim0_stride=256
      i32x8 g1 = {(int)(2u << 16), (int)(256u << 16), (int)(16u << 16),
                  (int)(256u << 16), 16, 256, 0, 0};
      i32x4 gz4 = {0, 0, 0, 0};
      i32x8 gz8 = {0, 0, 0, 0, 0, 0, 0, 0};
      // 6-arg (clang-23 / therock-10.0) form: (g0, g1, g2, g3, gX, cpol)
      __builtin_amdgcn_tensor_load_to_lds(g0, g1, gz4, gz4, gz8, 0);
    }
    __builtin_amdgcn_s_wait_tensorcnt(0);
  }
#else
  for (int i4 = tid; i4 < 4096; i4 += 512) {
    int base = i4 * 4;
    int nl = base >> 8, k = base & 255;
    int g = (nl >> 4) * H + w * 16 + (nl & 15);
    *(float4*)&sW[base] = *(const float4*)(Whh + (size_t)g * H + k);
  }
#endif
  for (int e = tid; e < 1024; e += 512)
    sC[e] = c0[(e >> 4) * H + w * 16 + (e & 15)];
  if (tid < 16) sWO[tid] = Wout[w * 16 + tid];
  __syncthreads();

  for (int s = 0; s < S; ++s) {
    const float* hsrc = hglob + (size_t)(s & 1) * (B * H);
    float* hdst = hglob + (size_t)((s & 1) ^ 1) * (B * H);

    // full h_{s-1} into LDS (64KB from L2) via async global->LDS B128 copies
    for (int i4 = tid; i4 < 4096; i4 += 512) {
      const float* gp = hsrc + i4 * 4;
      unsigned int la = (unsigned int)(size_t)&sH[i4 * 4];
      asm volatile("global_load_async_to_lds_b128 %0, %1, off"
                   :: "v"(la), "v"(gp) : "memory");
    }
    asm volatile("s_wait_asynccnt 0x0" ::: "memory");
    __syncthreads();

    // gates tile [16b x 16n]; accumulator seeded from precomputed x_gates
    v8f acc;
    {
      const float* xbase = xg + (size_t)s * B * G;
      int g = q * H + w * 16 + l15;
#pragma unroll
      for (int r = 0; r < 8; ++r) {
        int b = mb * 16 + r + 8 * half;
        acc[r] = xbase[(size_t)b * G + g];
      }
    }
    for (int k0 = 0; k0 < H; k0 += 4) {
      v2f af = *(const v2f*)&sH[(mb * 16 + l15) * H + k0 + 2 * half];
      v2f bf = *(const v2f*)&sW[(q * 16 + l15) * H + k0 + 2 * half];
      acc = WMMA_F32(af, bf, acc);
    }
#pragma unroll
    for (int r = 0; r < 8; ++r) {
      int b = mb * 16 + r + 8 * half;
      sGT[b * 64 + q * 16 + l15] = acc[r];
    }
    __syncthreads();

    // LSTM cell for this WG's 16 columns (1024 elems over 512 threads)
#pragma unroll
    for (int p = 0; p < 2; ++p) {
      int e = tid + p * 512;
      int b = e >> 4, j = e & 15;
      float iv = sigmoidf_(sGT[b * 64 + j]);
      float fv = sigmoidf_(sGT[b * 64 + 16 + j]);
      float gv = tanhf(sGT[b * 64 + 32 + j]);
      float ov = sigmoidf_(sGT[b * 64 + 48 + j]);
      float c = fv * sC[e] + iv * gv;
      sC[e] = c;
      float h = ov * tanhf(c);
      hdst[b * H + w * 16 + j] = h;  // publish h slice
      sHS[e] = h * sWO[j];           // partial output dot
    }
    __syncthreads();
    if (tid < 64) {
      float ysum = 0.0f;
#pragma unroll
      for (int j = 0; j < 16; ++j) ysum += sHS[tid * 16 + j];
      ypart[((size_t)s * 16 + w) * B + tid] = ysum;
    }

    __threadfence();  // release: h slice visible device-wide
    __syncthreads();  // all waves of this WG done before signaling
    if (tid < 32)     // one elected wave signals the cluster barrier
      asm volatile("s_barrier_signal -3" ::: "memory");
    asm volatile("s_barrier_wait -3" ::: "memory");
    __threadfence();  // acquire: drop stale h from near caches
  }
}

// ---------------------------------------------------------------------------
// Phase C: y[s,b] = sigmoid(sum_w ypart[s][w][b] + b_out)
// ---------------------------------------------------------------------------
__global__ void yfinal_kernel(const float* __restrict__ ypart,
                              const float* __restrict__ bout,
                              float* __restrict__ out) {
  int idx = blockIdx.x * blockDim.x + threadIdx.x;
  if (idx >= S * B) return;
  int s = idx >> 6, b = idx & 63;
  float a = bout[0];
#pragma unroll
  for (int w = 0; w < 16; ++w) a += ypart[((size_t)s * 16 + w) * B + b];
  out[idx] = 1.0f / (1.0f + __expf(-a));
}

// ---------------------------------------------------------------------------
extern "C" void kernel_launch(void* const* d_in, const int* in_sizes, int n_in,
                              void* d_out, int out_size, void* d_ws,
                              size_t ws_size, hipStream_t stream) {
  (void)in_sizes; (void)n_in; (void)out_size; (void)ws_size;
  const float* pw   = (const float*)d_in[0];
  const float* Wemb = (const float*)d_in[1];
  const float* bemb = (const float*)d_in[2];
  const float* Wih  = (const float*)d_in[3];
  const float* Whh  = (const float*)d_in[4];
  const float* bih  = (const float*)d_in[5];
  const float* bhh  = (const float*)d_in[6];
  const float* Wout = (const float*)d_in[7];
  const float* bout = (const float*)d_in[8];
  const float* h0   = (const float*)d_in[9];
  const float* c0   = (const float*)d_in[10];
  float* out = (float*)d_out;

  float* xg    = (float*)d_ws;               // S*B*G fp32 (512 MB)
  float* hglob = xg + (size_t)S * B * G;     // 2 * B*H (ping-pong h)
  float* ypart = hglob + 2 * B * H;          // S*16*B partial outputs

  (void)hipMemcpyAsync(hglob, h0, (size_t)B * H * sizeof(float),
                       hipMemcpyDeviceToDevice, stream);

  xgates_kernel<<<S, 512, 0, stream>>>(pw, Wemb, bemb, Wih, bih, bhh, xg);
  lstm_kernel<<<16, 512, 0, stream>>>(xg, Whh, Wout, c0, hglob, ypart);
  yfinal_kernel<<<(S * B + 255) / 256, 256, 0, stream>>>(ypart, bout, out);
}